// AttentionRNN_29463475651459
// MI455X (gfx1250) — compile-verified
//
#include <hip/hip_runtime.h>
#include <math.h>
#include <stdint.h>

// Problem constants (match reference)
#define Bb 128
#define Ss 1024
#define Hh 256
#define Vv 256
#define LDA 260   // 16-row LDS tiles padded: 260 % 64 == 4 -> conflict-free float2 lanes

typedef __attribute__((ext_vector_type(2))) float v2f;
typedef __attribute__((ext_vector_type(8))) float v8f;

// D = A(16x4 f32) * B(4x16 f32) + C, fp32 WMMA
__device__ __forceinline__ v8f wmma_f32_k4(v2f a, v2f b, v8f c) {
    return __builtin_amdgcn_wmma_f32_16x16x4_f32(
        /*neg_a=*/false, a, /*neg_b=*/false, b,
        /*c_mod=*/(short)0, c, /*reuse_a=*/false, /*reuse_b=*/false);
}

// Async global->LDS copy, 16B per lane (ASYNCcnt-tracked).
// lds_dst is a pointer into the __shared__ tile: ptrtoint makes the LDS object
// escape (so the "memory" clobber keeps it alive / ordered), and hardware maps
// flat LDS-aperture addresses by taking their low 32 bits, which is what the
// instruction's VDST operand consumes.
__device__ __forceinline__ void async_copy_b128(void* lds_dst, const void* gsrc) {
    unsigned lds_off = (unsigned)(uintptr_t)lds_dst;
    asm volatile("global_load_async_to_lds_b128 %0, %1, off"
                 :: "v"(lds_off), "v"(gsrc)
                 : "memory");
}
__device__ __forceinline__ void wait_asynccnt0() {
    asm volatile("s_wait_asynccnt 0x0" ::: "memory");
}

// ---------------------------------------------------------------------------
// Kernel 1: xg[s*B+b][n] = embed[x[b][s]] @ Wxh^T + (bxh + bhh)
// Grid: (B*S/16) workgroups x 512 threads (16 waves; wave w -> N-tile n0=16w).
// Wave w async-stages gathered A-tile row w into LDS (2 x b128 per lane-set).
// ---------------------------------------------------------------------------
__global__ __launch_bounds__(512) void k_embed_xgemm(
    const int* __restrict__ x, const float* __restrict__ embed,
    const float* __restrict__ Wxh, const float* __restrict__ bxh,
    const float* __restrict__ bhh, float* __restrict__ xg)
{
    __shared__ float aTile[16 * LDA];
    const int mTile = blockIdx.x;
    const int tid = threadIdx.x, lane = tid & 31, wave = tid >> 5;

    // Async gather: row w = embed[x[b][s]] (1024 B) -> LDS row w (stride 1040 B)
    {
        int m = mTile * 16 + wave;
        int b = m & (Bb - 1);
        int s = m >> 7;
        int tok = x[b * Ss + s];
        const char* src = (const char*)(embed + tok * Hh);
        char* dst = (char*)(aTile + wave * LDA);
        async_copy_b128(dst + lane * 16, src + lane * 16);
        async_copy_b128(dst + 512 + lane * 16, src + 512 + lane * 16);
    }
    wait_asynccnt0();
    __syncthreads();

    const int n0 = wave * 16, nl = lane & 15, hi = lane >> 4;
    const float bb = bxh[n0 + nl] + bhh[n0 + nl];
    v8f acc;
#pragma unroll
    for (int j = 0; j < 8; ++j) acc[j] = bb;

    const float* brow = Wxh + (n0 + nl) * Hh;   // B[k][n] = Wxh[n][k]
    const float* arow = aTile + nl * LDA;
#pragma unroll 8
    for (int kk = 0; kk < Hh / 4; ++kk) {
        int kb = kk * 4 + hi * 2;
        v2f a = *(const v2f*)(arow + kb);
        v2f bm = *(const v2f*)(brow + kb);
        acc = wmma_f32_k4(a, bm, acc);
    }

    const int mBase = mTile * 16;
#pragma unroll
    for (int j = 0; j < 8; ++j) {
        int m = mBase + j + hi * 8;
        xg[m * Hh + n0 + nl] = acc[j];
    }
}

// ---------------------------------------------------------------------------
// Kernel 2: strict recurrence h_t = tanh(xg_t + h_{t-1} @ Whh^T)
// Grid: 8 workgroups (16-row batch tiles) x 512 threads. h double-buffered in
// LDS. 4 independent WMMA accumulation chains shorten the per-step dependent
// chain; next step's xg tile is prefetched (global_prefetch_b8).
// ---------------------------------------------------------------------------
__global__ __launch_bounds__(512) void k_rnn(
    const float* __restrict__ xg, const float* __restrict__ Whh,
    float* __restrict__ hs)
{
    __shared__ float hbuf[2][16 * LDA];
    const int b0 = blockIdx.x * 16;
    const int tid = threadIdx.x, lane = tid & 31, wave = tid >> 5;
    const int n0 = wave * 16, nl = lane & 15, hi = lane >> 4;

    for (int idx = tid; idx < 16 * LDA; idx += 512) hbuf[0][idx] = 0.0f;
    __syncthreads();

    const float* brow = Whh + (n0 + nl) * Hh;
    int cur = 0;
    for (int t = 0; t < Ss; ++t) {
        const float* xgt = xg + (t * Bb + b0) * Hh;
        v8f a0, a1, a2, a3;
#pragma unroll
        for (int j = 0; j < 8; ++j) {
            a0[j] = xgt[(j + hi * 8) * Hh + n0 + nl];
            a1[j] = 0.0f; a2[j] = 0.0f; a3[j] = 0.0f;
        }

        if (t + 1 < Ss) {   // pull next step's accumulator-init tile toward L0
            const float* nxt = xg + ((t + 1) * Bb + b0) * Hh;
            __builtin_prefetch(nxt + tid * 8, 0, 3);
        }

        const float* hrow = hbuf[cur] + nl * LDA;
#pragma unroll 4
        for (int kk = 0; kk < Hh / 4; kk += 4) {
            int kb = kk * 4 + hi * 2;
            a0 = wmma_f32_k4(*(const v2f*)(hrow + kb),      *(const v2f*)(brow + kb),      a0);
            a1 = wmma_f32_k4(*(const v2f*)(hrow + kb + 4),  *(const v2f*)(brow + kb + 4),  a1);
            a2 = wmma_f32_k4(*(const v2f*)(hrow + kb + 8),  *(const v2f*)(brow + kb + 8),  a2);
            a3 = wmma_f32_k4(*(const v2f*)(hrow + kb + 12), *(const v2f*)(brow + kb + 12), a3);
        }
        v8f acc = (a0 + a1) + (a2 + a3);

        float* hst = hs + (t * Bb + b0) * Hh;
        float* hnxt = hbuf[cur ^ 1];
#pragma unroll
        for (int j = 0; j < 8; ++j) {
            float v = tanhf(acc[j]);
            int r = j + hi * 8;
            hst[r * Hh + n0 + nl] = v;
            hnxt[r * LDA + n0 + nl] = v;
        }
        __syncthreads();
        cur ^= 1;
    }
}

// ---------------------------------------------------------------------------
// Kernel 3: scores = hs @ Wattn^T + battn   (hs rows contiguous: m = s*B + b)
// A tile staged via async global->LDS copy.
// ---------------------------------------------------------------------------
__global__ __launch_bounds__(512) void k_scores(
    const float* __restrict__ hs, const float* __restrict__ Wattn,
    const float* __restrict__ battn, float* __restrict__ scores)
{
    __shared__ float aTile[16 * LDA];
    const int mTile = blockIdx.x;
    const int tid = threadIdx.x, lane = tid & 31, wave = tid >> 5;

    {
        const char* src = (const char*)(hs + (mTile * 16 + wave) * Hh);
        char* dst = (char*)(aTile + wave * LDA);
        async_copy_b128(dst + lane * 16, src + lane * 16);
        async_copy_b128(dst + 512 + lane * 16, src + 512 + lane * 16);
    }
    wait_asynccnt0();
    __syncthreads();

    const int n0 = wave * 16, nl = lane & 15, hi = lane >> 4;
    const float bb = battn[n0 + nl];
    v8f acc;
#pragma unroll
    for (int j = 0; j < 8; ++j) acc[j] = bb;

    const float* brow = Wattn + (n0 + nl) * Hh;
    const float* arow = aTile + nl * LDA;
#pragma unroll 8
    for (int kk = 0; kk < Hh / 4; ++kk) {
        int kb = kk * 4 + hi * 2;
        v2f a = *(const v2f*)(arow + kb);
        v2f bm = *(const v2f*)(brow + kb);
        acc = wmma_f32_k4(a, bm, acc);
    }

    const int mBase = mTile * 16;
#pragma unroll
    for (int j = 0; j < 8; ++j) {
        int m = mBase + j + hi * 8;
        scores[m * Hh + n0 + nl] = acc[j];
    }
}

// ---------------------------------------------------------------------------
// Kernel 4: online softmax over S per (b,n) + context = sum(w * hs)
// Grid: B workgroups x 256 threads (thread n owns feature column n)
// ---------------------------------------------------------------------------
__global__ __launch_bounds__(256) void k_softmax_ctx(
    const float* __restrict__ scores, const float* __restrict__ hs,
    float* __restrict__ context)
{
    const int b = blockIdx.x, n = threadIdx.x;
    float m = -INFINITY, Z = 0.0f, ctx = 0.0f;
    for (int s = 0; s < Ss; ++s) {
        int off = (s * Bb + b) * Hh + n;
        float v = scores[off];
        float h = hs[off];
        float mn = fmaxf(m, v);
        float corr = __expf(m - mn);
        float e = __expf(v - mn);
        Z = Z * corr + e;
        ctx = ctx * corr + e * h;
        m = mn;
    }
    context[b * Hh + n] = ctx / Z;
}

// ---------------------------------------------------------------------------
// Kernel 5: out[b][s][v] = (hs + context[b]) @ Wfc^T + bfc
// hs tile staged async; +context applied in-LDS between barriers.
// ---------------------------------------------------------------------------
__global__ __launch_bounds__(512) void k_out(
    const float* __restrict__ hs, const float* __restrict__ context,
    const float* __restrict__ Wfc, const float* __restrict__ bfc,
    float* __restrict__ out)
{
    __shared__ float aTile[16 * LDA];
    const int mTile = blockIdx.x;
    const int tid = threadIdx.x, lane = tid & 31, wave = tid >> 5;

    {
        const char* src = (const char*)(hs + (mTile * 16 + wave) * Hh);
        char* dst = (char*)(aTile + wave * LDA);
        async_copy_b128(dst + lane * 16, src + lane * 16);
        async_copy_b128(dst + 512 + lane * 16, src + 512 + lane * 16);
    }
    wait_asynccnt0();
    __syncthreads();

    // in-LDS fixup: aTile[r][c] += context[b(r)][c]
    for (int idx = tid; idx < 16 * Hh; idx += 512) {
        int r = idx >> 8, c = idx & 255;
        int b = (mTile * 16 + r) & (Bb - 1);
        aTile[r * LDA + c] += context[b * Hh + c];
    }
    __syncthreads();

    const int n0 = wave * 16, nl = lane & 15, hi = lane >> 4;
    const float bb = bfc[n0 + nl];
    v8f acc;
#pragma unroll
    for (int j = 0; j < 8; ++j) acc[j] = bb;

    const float* brow = Wfc + (n0 + nl) * Hh;
    const float* arow = aTile + nl * LDA;
#pragma unroll 8
    for (int kk = 0; kk < Hh / 4; ++kk) {
        int kb = kk * 4 + hi * 2;
        v2f a = *(const v2f*)(arow + kb);
        v2f bm = *(const v2f*)(brow + kb);
        acc = wmma_f32_k4(a, bm, acc);
    }

    const int mBase = mTile * 16;
#pragma unroll
    for (int j = 0; j < 8; ++j) {
        int m = mBase + j + hi * 8;
        int b = m & (Bb - 1);
        int s = m >> 7;
        out[((b * Ss) + s) * Vv + n0 + nl] = acc[j];
    }
}

// ---------------------------------------------------------------------------
extern "C" void kernel_launch(void* const* d_in, const int* in_sizes, int n_in,
                              void* d_out, int out_size, void* d_ws, size_t ws_size,
                              hipStream_t stream)
{
    (void)in_sizes; (void)n_in; (void)out_size;
    const int*   x     = (const int*)d_in[0];
    const float* embed = (const float*)d_in[1];
    const float* Wxh   = (const float*)d_in[2];
    const float* bxh   = (const float*)d_in[3];
    const float* Whh   = (const float*)d_in[4];
    const float* bhh   = (const float*)d_in[5];
    const float* Wattn = (const float*)d_in[6];
    const float* battn = (const float*)d_in[7];
    const float* Wfc   = (const float*)d_in[8];
    const float* bfc   = (const float*)d_in[9];
    float* out = (float*)d_out;

    const size_t bshBytes = (size_t)Bb * Ss * Hh * sizeof(float); // 128 MiB
    if (ws_size < 2 * bshBytes + (size_t)Bb * Hh * sizeof(float)) return;

    char* ws = (char*)d_ws;
    float* hs  = (float*)ws;                          // [S,B,H]
    float* xg  = (float*)(ws + bshBytes);             // [S,B,H]; reused as scores
    float* ctx = (float*)(ws + 2 * bshBytes);         // [B,H]

    const int mTiles = (Bb * Ss) / 16;                // 8192
    k_embed_xgemm<<<mTiles, 512, 0, stream>>>(x, embed, Wxh, bxh, bhh, xg);
    k_rnn<<<Bb / 16, 512, 0, stream>>>(xg, Whh, hs);
    k_scores<<<mTiles, 512, 0, stream>>>(hs, Wattn, battn, xg /*scores*/);
    k_softmax_ctx<<<Bb, 256, 0, stream>>>(xg, hs, ctx);
    k_out<<<mTiles, 512, 0, stream>>>(hs, ctx, Wfc, bfc, out);
}